// BaselineModel_65317862637682
// MI455X (gfx1250) — compile-verified
//
#include <hip/hip_runtime.h>

typedef float v2f __attribute__((ext_vector_type(2)));
typedef float v8f __attribute__((ext_vector_type(8)));

#define HIDDEN   8
#define NGRAPH   16
#define LIN_IN   80000
#define NOUT1    1000
#define NTILES   63      // ceil(1000/16)
#define KCH      50      // k-chunks for lin1
#define KSTEP    1600    // 80000 / 50
#define TB       256

// ---------------- utility kernels ----------------
__global__ void k_zero(float* p, int n) {
    int i = blockIdx.x * blockDim.x + threadIdx.x;
    if (i < n) p[i] = 0.0f;
}

__global__ void k_copy(const float* a, float* b, int n) {
    int i = blockIdx.x * blockDim.x + threadIdx.x;
    if (i < n) b[i] = a[i];
}

// ---------------- normalization ----------------
__global__ void k_deg(const int* src, const int* dst, float* deg, int E) {
    int e = blockIdx.x * blockDim.x + threadIdx.x;
    if (e < E) {
        int s = src[e], d = dst[e];
        if (s != d) atomicAdd(&deg[s], 1.0f);
    }
}

__global__ void k_rsqrt(float* deg, int n) {
    int i = blockIdx.x * blockDim.x + threadIdx.x;
    if (i < n) {
        float d = deg[i];
        deg[i] = (d > 0.0f) ? rsqrtf(fmaxf(d, 1.0f)) : 0.0f;
    }
}

__global__ void k_norm(const int* src, const int* dst, const float* dis,
                       float* nrm, int E) {
    int e = blockIdx.x * blockDim.x + threadIdx.x;
    if (e < E) {
        int s = src[e], d = dst[e];
        nrm[e] = (s != d) ? (-dis[s] * dis[d]) : 0.0f;
    }
}

// ---------------- conv1 (C = 1) ----------------
__global__ void k_prop1(const int* src, const int* dst, const float* nrm,
                        const float* tin, float* tout, int E) {
    int e = blockIdx.x * blockDim.x + threadIdx.x;
    if (e < E) {
        float w = nrm[e];
        if (w != 0.0f) atomicAdd(&tout[dst[e]], w * tin[src[e]]);
    }
}

__global__ void k_combine(float* t2, const float* t0, int n) {
    int i = blockIdx.x * blockDim.x + threadIdx.x;
    if (i < n) t2[i] = 2.0f * t2[i] - t0[i];
}

// out[n][c] (+)= t[n] * W[c]   (W = conv1_w + k*8)
__global__ void k_acc1(const float* t, const float* W, float* out, int total,
                       int init) {
    int i = blockIdx.x * blockDim.x + threadIdx.x;
    if (i < total) {
        int n = i >> 3, c = i & 7;
        float v = t[n] * W[c];
        out[i] = init ? v : (out[i] + v);
    }
}

__global__ void k_bias_relu(float* h, const float* b, int C, int total) {
    int i = blockIdx.x * blockDim.x + threadIdx.x;
    if (i < total) h[i] = fmaxf(h[i] + b[i % C], 0.0f);
}

// ---------------- conv2 (C = 8) ----------------
__global__ void k_prop8(const int* src, const int* dst, const float* nrm,
                        const float* uin, float* uout, int E) {
    int e = blockIdx.x * blockDim.x + threadIdx.x;
    if (e < E) {
        float w = nrm[e];
        if (w == 0.0f) return;
        int s = src[e], d = dst[e];
        const float4* pin = (const float4*)(uin + (size_t)s * 8);
        float4 lo = pin[0], hi = pin[1];
        float* po = uout + (size_t)d * 8;
        atomicAdd(po + 0, w * lo.x);
        atomicAdd(po + 1, w * lo.y);
        atomicAdd(po + 2, w * lo.z);
        atomicAdd(po + 3, w * lo.w);
        atomicAdd(po + 4, w * hi.x);
        atomicAdd(po + 5, w * hi.y);
        atomicAdd(po + 6, w * hi.z);
        atomicAdd(po + 7, w * hi.w);
    }
}

// out[n][:] (+)= u[n][:] @ W   (W = conv2_w + k*64, [8][8] row-major)
__global__ void k_acc8(const float* u, const float* W, float* out, int N,
                       int init) {
    int n = blockIdx.x * blockDim.x + threadIdx.x;
    if (n < N) {
        const float4* pu = (const float4*)(u + (size_t)n * 8);
        float4 lo = pu[0], hi = pu[1];
        float in[8] = {lo.x, lo.y, lo.z, lo.w, hi.x, hi.y, hi.z, hi.w};
        float* po = out + (size_t)n * 8;
#pragma unroll
        for (int c = 0; c < 8; ++c) {
            float s = init ? 0.0f : po[c];
#pragma unroll
            for (int ci = 0; ci < 8; ++ci) s += in[ci] * W[ci * 8 + c];
            po[c] = s;
        }
    }
}

// ---------------- lin1 via V_WMMA_F32_16X16X4_F32 ----------------
// A = h2 viewed as [16][80000] row-major, W = lin1_w [80000][1000].
// One wave per (16-col tile, K-chunk); f32 atomics combine K-chunk partials.
__global__ __launch_bounds__(32) void k_lin1_wmma(const float* __restrict__ A,
                                                  const float* __restrict__ W,
                                                  float* __restrict__ acc) {
    const int tile = blockIdx.x;          // 0..62
    const int kch  = blockIdx.y;          // 0..KCH-1
    const int lane = threadIdx.x;         // 0..31
    const int n0   = tile * 16;
    const int half = lane >> 4;           // 0: K=0,1   1: K=2,3
    const int mrow = lane & 15;           // A row for this lane
    int colc = n0 + (lane & 15);          // clamped column for B loads
    if (colc > NOUT1 - 1) colc = NOUT1 - 1;

    const float* Arow = A + (size_t)mrow * LIN_IN;
    const int kBeg = kch * KSTEP;
    const int kEnd = kBeg + KSTEP;

    v8f c = {};
    for (int k = kBeg; k < kEnd; k += 4) {
        const int ka = k + half * 2;
        v2f a, b;
        a.x = Arow[ka];
        a.y = Arow[ka + 1];
        b.x = W[(size_t)ka * NOUT1 + colc];
        b.y = W[(size_t)(ka + 1) * NOUT1 + colc];
        c = __builtin_amdgcn_wmma_f32_16x16x4_f32(
            /*neg_a=*/false, a, /*neg_b=*/false, b,
            /*c_mod=*/(short)0, c, /*reuse_a=*/false, /*reuse_b=*/false);
    }

    const int nc = n0 + (lane & 15);
    if (nc < NOUT1) {
#pragma unroll
        for (int r = 0; r < 8; ++r) {
            int m = (lane < 16) ? r : (r + 8);
            atomicAdd(&acc[(size_t)m * NOUT1 + nc], c[r]);
        }
    }
}

// ---------------- lin2 + clip ----------------
__global__ void k_lin2(const float* __restrict__ h, const float* __restrict__ w,
                       const float* __restrict__ b, float* __restrict__ out) {
    __shared__ float red[256];
    const int g = blockIdx.x;
    const int t = threadIdx.x;
    float s = 0.0f;
    for (int i = t; i < NOUT1; i += 256) s += h[(size_t)g * NOUT1 + i] * w[i];
    red[t] = s;
    __syncthreads();
    for (int o = 128; o > 0; o >>= 1) {
        if (t < o) red[t] += red[t + o];
        __syncthreads();
    }
    if (t == 0) {
        float v = red[0] + b[0];
        out[g] = fminf(fmaxf(v, 0.0f), 110.0f);
    }
}

// ---------------- host ----------------
static inline int cdiv(int a, int b) { return (a + b - 1) / b; }

extern "C" void kernel_launch(void* const* d_in, const int* in_sizes, int n_in,
                              void* d_out, int out_size, void* d_ws, size_t ws_size,
                              hipStream_t stream) {
    const float* x   = (const float*)d_in[0];
    const int*   ei  = (const int*)d_in[1];
    // d_in[2] edge_attr unused, d_in[3] batch unused
    const float* W1  = (const float*)d_in[4];   // [5,1,8]
    const float* b1  = (const float*)d_in[5];
    const float* W2  = (const float*)d_in[6];   // [5,8,8]
    const float* b2  = (const float*)d_in[7];
    const float* L1w = (const float*)d_in[8];   // [80000,1000]
    const float* L1b = (const float*)d_in[9];
    const float* L2w = (const float*)d_in[10];  // [1000,1]
    const float* L2b = (const float*)d_in[11];

    const int N = in_sizes[0];          // 160000
    const int E = in_sizes[1] / 2;      // 4,000,000
    const int* src = ei;
    const int* dst = ei + E;

    // --- carve workspace ---
    char* w = (char*)d_ws;
    auto carve = [&](size_t bytes) -> float* {
        float* p = (float*)w;
        w += (bytes + 255) & ~(size_t)255;
        return p;
    };
    float* nrm  = carve((size_t)E * 4);          // 16 MB
    float* dis  = carve((size_t)N * 4);          // deg then rsqrt in place
    float* t0   = carve((size_t)N * 4);
    float* t1   = carve((size_t)N * 4);
    float* t2   = carve((size_t)N * 4);
    float* o1   = carve((size_t)N * 8 * 4);      // conv1 out -> h1 -> Tx0 of conv2
    float* uA   = carve((size_t)N * 8 * 4);
    float* uB   = carve((size_t)N * 8 * 4);
    float* o2   = carve((size_t)N * 8 * 4);      // conv2 out -> h2 = A[16][80000]
    float* acc1 = carve((size_t)NGRAPH * NOUT1 * 4);

    const int gE  = cdiv(E, TB);
    const int gN  = cdiv(N, TB);
    const int gN8 = cdiv(N * 8, TB);

    // --- normalization ---
    k_zero<<<gN, TB, 0, stream>>>(dis, N);
    k_deg<<<gE, TB, 0, stream>>>(src, dst, dis, E);
    k_rsqrt<<<gN, TB, 0, stream>>>(dis, N);
    k_norm<<<gE, TB, 0, stream>>>(src, dst, dis, nrm, E);

    // --- conv1: C=1 Chebyshev recursion ---
    k_copy<<<gN, TB, 0, stream>>>(x, t0, N);
    k_acc1<<<gN8, TB, 0, stream>>>(t0, W1 + 0, o1, N * 8, 1);
    k_zero<<<gN, TB, 0, stream>>>(t1, N);
    k_prop1<<<gE, TB, 0, stream>>>(src, dst, nrm, t0, t1, E);
    k_acc1<<<gN8, TB, 0, stream>>>(t1, W1 + 8, o1, N * 8, 0);
    {
        float *p0 = t0, *p1 = t1, *p2 = t2;
        for (int k = 2; k < 5; ++k) {
            k_zero<<<gN, TB, 0, stream>>>(p2, N);
            k_prop1<<<gE, TB, 0, stream>>>(src, dst, nrm, p1, p2, E);
            k_combine<<<gN, TB, 0, stream>>>(p2, p0, N);
            k_acc1<<<gN8, TB, 0, stream>>>(p2, W1 + k * 8, o1, N * 8, 0);
            float* tmp = p0; p0 = p1; p1 = p2; p2 = tmp;
        }
    }
    k_bias_relu<<<gN8, TB, 0, stream>>>(o1, b1, 8, N * 8);   // o1 = h1

    // --- conv2: C=8 Chebyshev recursion ---
    k_acc8<<<gN, TB, 0, stream>>>(o1, W2 + 0, o2, N, 1);
    k_zero<<<gN8, TB, 0, stream>>>(uA, N * 8);
    k_prop8<<<gE, TB, 0, stream>>>(src, dst, nrm, o1, uA, E);
    k_acc8<<<gN, TB, 0, stream>>>(uA, W2 + 64, o2, N, 0);
    {
        float *p0 = o1, *p1 = uA, *p2 = uB;
        for (int k = 2; k < 5; ++k) {
            k_zero<<<gN8, TB, 0, stream>>>(p2, N * 8);
            k_prop8<<<gE, TB, 0, stream>>>(src, dst, nrm, p1, p2, E);
            k_combine<<<gN8, TB, 0, stream>>>(p2, p0, N * 8);
            k_acc8<<<gN, TB, 0, stream>>>(p2, W2 + k * 64, o2, N, 0);
            float* tmp = p0; p0 = p1; p1 = p2; p2 = tmp;
        }
    }
    k_bias_relu<<<gN8, TB, 0, stream>>>(o2, b2, 8, N * 8);   // o2 = h2

    // --- lin1 via f32 WMMA ---
    k_zero<<<cdiv(NGRAPH * NOUT1, TB), TB, 0, stream>>>(acc1, NGRAPH * NOUT1);
    k_lin1_wmma<<<dim3(NTILES, KCH), 32, 0, stream>>>(o2, L1w, acc1);
    k_bias_relu<<<cdiv(NGRAPH * NOUT1, TB), TB, 0, stream>>>(acc1, L1b, NOUT1,
                                                             NGRAPH * NOUT1);

    // --- lin2 + clip ---
    k_lin2<<<NGRAPH, 256, 0, stream>>>(acc1, L2w, L2b, (float*)d_out);
}